// VMDPeriodDetection_80324478369859
// MI455X (gfx1250) — compile-verified
//
#include <hip/hip_runtime.h>
#include <math.h>

typedef __attribute__((ext_vector_type(2))) float v2f;
typedef __attribute__((ext_vector_type(8))) float v8f;

#define Bn 32
#define Tn 4096
#define Cn 64
#define Fn 2049          // T/2 + 1
#define Kn 2
#define ALPHAc 2000.0f
#define TOLc 1e-7f
#define MAXIT 30

// ---- workspace layout (float offsets) ----
#define SX 4196352       // 2048 * 2049  (one component of X, column-major n*F+f)
#define SU 8392704       // 4096 * 2049  (one component of u, (b*K+k)*C+c major)
#define OFF_XRE 0
#define OFF_XIM (SX)
#define OFF_URE (2*SX)
#define OFF_UIM (2*SX + SU)
#define OFF_SNP (2*SX + 2*SU)          // sum n*power  [B*K*C] = 4096
#define OFF_SP  (OFF_SNP + 4096)       // sum power    [B*K*C] = 4096
#define OFF_CONV (OFF_SP + 4096)       // conv partial [B*C]   = 2048
#define OFF_OMEGA (OFF_CONV + 2048)    // omega [B*K] = 64
#define OFF_TOP2 (OFF_OMEGA + 64)      // 64 * 6
#define OFF_FLAG (OFF_TOP2 + 384)      // done flag (int in float slot)
#define WS_FLOATS (OFF_FLAG + 64)

// ---------------- zero init (u, sums, omega, flag) ----------------
__global__ __launch_bounds__(256) void vmd_zero(float* __restrict__ ws) {
  const size_t n = (size_t)WS_FLOATS - (size_t)OFF_URE;
  float* p = ws + OFF_URE;
  for (size_t i = (size_t)blockIdx.x * 256 + threadIdx.x; i < n;
       i += (size_t)gridDim.x * 256)
    p[i] = 0.0f;
}

// ---------------- rfft as WMMA f32 GEMM ----------------
// X[f, n] = sum_t exp(-2*pi*i*f*t/T) * x[t, n],  n = b*64 + c
// M (f) padded 2049 -> 2064 (129 tiles), N = 2048 (128 tiles), K = 4096.
__global__ __launch_bounds__(256) void dft_wmma(const float* __restrict__ x,
                                                float* __restrict__ ws) {
  float* XRe = ws + OFF_XRE;
  float* XIm = ws + OFF_XIM;
  const int lane = threadIdx.x & 31;
  const int wv   = threadIdx.x >> 5;
  const int gw   = blockIdx.x * 8 + wv;      // 2064 blocks * 8 waves = 16512
  const int mt   = gw >> 7;                  // / 128 n-tiles
  const int nt   = gw & 127;
  if (mt >= 129) return;
  const int fbase = mt << 4;
  const int nbase = nt << 4;
  const int b  = nbase >> 6;
  const int c0 = nbase & 63;
  const float* __restrict__ xcol = x + (size_t)b * Tn * Cn + c0;
  const int l16 = lane & 15;
  const int lh  = lane >> 4;
  const int fA  = fbase + l16;               // A-matrix row (M) for this lane
  v8f accRe = {0.f,0.f,0.f,0.f,0.f,0.f,0.f,0.f};
  v8f accIm = {0.f,0.f,0.f,0.f,0.f,0.f,0.f,0.f};
  const float SCALE = -6.28318530717958647692f / 4096.0f;
  for (int k0 = 0; k0 < Tn; k0 += 4) {
    const int t0 = k0 + 2 * lh;
    const int t1 = t0 + 1;
    // exact integer phase, reduced mod T (f*t < 2^24, exact in int32)
    const int ph0 = (fA * t0) & 4095;
    const int ph1 = (fA * t1) & 4095;
    float s0, c0w, s1, c1w;
    __sincosf(SCALE * (float)ph0, &s0, &c0w);
    __sincosf(SCALE * (float)ph1, &s1, &c1w);
    v2f aRe; aRe.x = c0w; aRe.y = c1w;       // cos(2*pi*f*t/T)
    v2f aIm; aIm.x = s0;  aIm.y = s1;        // -sin(2*pi*f*t/T)
    v2f bX;
    bX.x = xcol[t0 * Cn + l16];
    bX.y = xcol[t1 * Cn + l16];
    __builtin_prefetch(&xcol[(t0 + 16) * Cn + l16], 0, 0);
    accRe = __builtin_amdgcn_wmma_f32_16x16x4_f32(
        false, aRe, false, bX, (short)0, accRe, false, false);
    accIm = __builtin_amdgcn_wmma_f32_16x16x4_f32(
        false, aIm, false, bX, (short)0, accIm, false, false);
  }
  const int ncol = nbase + l16;
#pragma unroll
  for (int j = 0; j < 8; ++j) {
    const int f = fbase + j + 8 * lh;        // D VGPR j: M = j (+8 for hi lanes)
    if (f < Fn) {
      XRe[(size_t)ncol * Fn + f] = accRe[j];
      XIm[(size_t)ncol * Fn + f] = accIm[j];
    }
  }
}

// ---------------- one VMD iteration: update u, partial sums ----------------
__global__ __launch_bounds__(256) void vmd_step(float* __restrict__ ws) {
  const int* flag = (const int*)(ws + OFF_FLAG);
  if (*flag) return;                         // frozen: keep u/omega as-is
  const int col = blockIdx.x;                // 0..2047  (b*64 + c)
  const int b = col >> 6;
  const int c = col & 63;
  const float* __restrict__ XRe = ws + OFF_XRE + (size_t)col * Fn;
  const float* __restrict__ XIm = ws + OFF_XIM + (size_t)col * Fn;
  float* __restrict__ uRe0 = ws + OFF_URE + ((size_t)(b * 2 + 0) * 64 + c) * Fn;
  float* __restrict__ uIm0 = ws + OFF_UIM + ((size_t)(b * 2 + 0) * 64 + c) * Fn;
  float* __restrict__ uRe1 = ws + OFF_URE + ((size_t)(b * 2 + 1) * 64 + c) * Fn;
  float* __restrict__ uIm1 = ws + OFF_UIM + ((size_t)(b * 2 + 1) * 64 + c) * Fn;
  const float w0 = ws[OFF_OMEGA + 2 * b + 0];
  const float w1 = ws[OFF_OMEGA + 2 * b + 1];
  float np0 = 0.f, p0 = 0.f, np1 = 0.f, p1 = 0.f, cv = 0.f;
  for (int f = threadIdx.x; f < Fn; f += 256) {
    const float xr = XRe[f], xi = XIm[f];
    const float a0r = uRe0[f], a0i = uIm0[f];
    const float a1r = uRe1[f], a1i = uIm1[f];
    const float nr = xr - a0r - a1r;         // numer (lam == 0 since TAU == 0)
    const float ni = xi - a0i - a1i;
    const float nf = (float)(f + 1);         // n = arange(1, F+1)
    float d0 = nf - w0; d0 = 1.0f + ALPHAc * d0 * d0;
    float d1 = nf - w1; d1 = 1.0f + ALPHAc * d1 * d1;
    const float i0 = 1.0f / d0, i1 = 1.0f / d1;
    const float u0r = nr * i0, u0i = ni * i0;
    const float u1r = nr * i1, u1i = ni * i1;
    const float pw0 = u0r * u0r + u0i * u0i;
    const float pw1 = u1r * u1r + u1i * u1i;
    np0 += nf * pw0; p0 += pw0;
    np1 += nf * pw1; p1 += pw1;
    const float dr0 = u0r - a0r, di0 = u0i - a0i;
    const float dr1 = u1r - a1r, di1 = u1i - a1i;
    cv += dr0 * dr0 + di0 * di0 + dr1 * dr1 + di1 * di1;
    uRe0[f] = u0r; uIm0[f] = u0i;
    uRe1[f] = u1r; uIm1[f] = u1i;
  }
  __shared__ float s0[256], s1[256], s2[256], s3[256], s4[256];
  const int t = threadIdx.x;
  s0[t] = np0; s1[t] = p0; s2[t] = np1; s3[t] = p1; s4[t] = cv;
  __syncthreads();
  for (int st = 128; st > 0; st >>= 1) {
    if (t < st) {
      s0[t] += s0[t + st]; s1[t] += s1[t + st]; s2[t] += s2[t + st];
      s3[t] += s3[t + st]; s4[t] += s4[t + st];
    }
    __syncthreads();
  }
  if (t == 0) {
    ws[OFF_SNP + (2 * b + 0) * 64 + c] = s0[0];
    ws[OFF_SP  + (2 * b + 0) * 64 + c] = s1[0];
    ws[OFF_SNP + (2 * b + 1) * 64 + c] = s2[0];
    ws[OFF_SP  + (2 * b + 1) * 64 + c] = s3[0];
    ws[OFF_CONV + col] = s4[0];
  }
}

// ---------------- omega update + convergence flag ----------------
__global__ __launch_bounds__(256) void vmd_omega(float* __restrict__ ws, int it) {
  int* flag = (int*)(ws + OFF_FLAG);
  const int d = *flag;
  const int t = threadIdx.x;
  if (!d && t < 64) {                        // t = b*2 + k
    const float* snp = ws + OFF_SNP + t * 64;
    const float* sp  = ws + OFF_SP  + t * 64;
    float r = 0.f;
    for (int c = 0; c < 64; ++c) r += snp[c] / (sp[c] + 1e-10f);
    ws[OFF_OMEGA + t] = r * (1.0f / 64.0f);  // mean over C
  }
  float s = 0.f;
  for (int i = t; i < 2048; i += 256) s += ws[OFF_CONV + i];
  __shared__ float red[256];
  red[t] = s; __syncthreads();
  for (int st = 128; st > 0; st >>= 1) {
    if (t < st) red[t] += red[t + st];
    __syncthreads();
  }
  if (t == 0 && !d && ((it % 5) == 4)) {
    const float conv = red[0] / 8392704.0f;  // mean over B*K*F*C
    if (conv < TOLc) *flag = 1;
  }
}

// ---------------- top-2 of fl over F per (b, mode) ----------------
__device__ __forceinline__ bool better2(float a, int ia, float b, int ib) {
  return (a > b) || (a == b && ia < ib);
}
__device__ __forceinline__ void insert2(float w, int j, float s,
                                        float& v0, int& i0, float& r0,
                                        float& v1, int& i1, float& r1) {
  if (better2(w, j, v0, i0)) { v1 = v0; i1 = i0; r1 = r0; v0 = w; i0 = j; r0 = s; }
  else if (better2(w, j, v1, i1)) { v1 = w; i1 = j; r1 = s; }
}

__global__ __launch_bounds__(256) void vmd_topk(float* __restrict__ ws) {
  const int bk = blockIdx.x;                 // 0..63 = b*2 + mode
  const float* uRe = ws + OFF_URE + (size_t)bk * 64 * Fn;
  const float* uIm = ws + OFF_UIM + (size_t)bk * 64 * Fn;
  float v0 = -1.f, v1 = -1.f, r0 = 0.f, r1 = 0.f;
  int i0 = 1 << 30, i1 = 1 << 30;
  for (int f = threadIdx.x; f < Fn; f += 256) {
    float s = 0.f;
    const bool edge = (f == 0) || (f == Fn - 1);   // DC/Nyquist: imag dropped
    for (int c = 0; c < 64; ++c) {
      const float re = uRe[(size_t)c * Fn + f];
      const float im = uIm[(size_t)c * Fn + f];
      s += edge ? fabsf(re) : sqrtf(re * re + im * im);
    }
    const float fl  = s * (1.0f / 64.0f);    // fl_raw (mean over C)
    const float key = (f == 0) ? 0.0f : fl;  // fl[...,0] = 0
    insert2(key, f, fl, v0, i0, r0, v1, i1, r1);
  }
  __shared__ float sv0[256], sr0[256], sv1[256], sr1[256];
  __shared__ int   si0[256], si1[256];
  const int t = threadIdx.x;
  sv0[t] = v0; si0[t] = i0; sr0[t] = r0; sv1[t] = v1; si1[t] = i1; sr1[t] = r1;
  __syncthreads();
  for (int st = 128; st > 0; st >>= 1) {
    if (t < st) {
      float w0 = sv0[t], w1 = sv1[t], q0 = sr0[t], q1 = sr1[t];
      int j0 = si0[t], j1 = si1[t];
      insert2(sv0[t + st], si0[t + st], sr0[t + st], w0, j0, q0, w1, j1, q1);
      insert2(sv1[t + st], si1[t + st], sr1[t + st], w0, j0, q0, w1, j1, q1);
      sv0[t] = w0; si0[t] = j0; sr0[t] = q0; sv1[t] = w1; si1[t] = j1; sr1[t] = q1;
    }
    __syncthreads();
  }
  if (t == 0) {
    float* o = ws + OFF_TOP2 + bk * 6;
    o[0] = sv0[0]; o[1] = (float)si0[0]; o[2] = sr0[0];
    o[3] = sv1[0]; o[4] = (float)si1[0]; o[5] = sr1[0];
  }
}

// ---------------- final: periods (2) + evals (32x2) ----------------
__global__ void vmd_final(const float* __restrict__ ws, float* __restrict__ out) {
  __shared__ float sp0[32], sp1[32];
  const int b = threadIdx.x;
  if (b < 32) {
    float e[2], per[2];
    for (int km = 0; km < 2; ++km) {
      const float* o = ws + OFF_TOP2 + (b * 2 + km) * 6;
      const float raw0 = o[2], raw1 = o[5];
      int idx1 = (int)o[4]; if (idx1 < 1) idx1 = 1;
      e[km]   = 0.5f * (raw0 + raw1);        // energies: mean of top-2 fl_raw
      per[km] = (float)(Tn / idx1);          // T // idx[..., k-1]
    }
    const int first  = (e[0] >= e[1]) ? 0 : 1;   // top_k over K, ties -> lower
    const int second = 1 - first;
    out[2 + 2 * b + 0] = e[first];
    out[2 + 2 * b + 1] = e[second];
    sp0[b] = per[first]; sp1[b] = per[second];
  }
  __syncthreads();
  if (b == 0) {
    float a = 0.f, bb = 0.f;
    for (int i = 0; i < 32; ++i) { a += sp0[i]; bb += sp1[i]; }
    out[0] = a * (1.0f / 32.0f);
    out[1] = bb * (1.0f / 32.0f);
  }
}

extern "C" void kernel_launch(void* const* d_in, const int* in_sizes, int n_in,
                              void* d_out, int out_size, void* d_ws, size_t ws_size,
                              hipStream_t stream) {
  const float* x = (const float*)d_in[0];
  float* out = (float*)d_out;
  float* ws  = (float*)d_ws;
  if (ws_size < (size_t)WS_FLOATS * sizeof(float)) return;  // need ~101 MB

  hipLaunchKernelGGL(vmd_zero, dim3(4096), dim3(256), 0, stream, ws);
  hipLaunchKernelGGL(dft_wmma, dim3(2064), dim3(256), 0, stream, x, ws);
  for (int it = 0; it < MAXIT; ++it) {
    hipLaunchKernelGGL(vmd_step,  dim3(2048), dim3(256), 0, stream, ws);
    hipLaunchKernelGGL(vmd_omega, dim3(1),    dim3(256), 0, stream, ws, it);
  }
  hipLaunchKernelGGL(vmd_topk,  dim3(64), dim3(256), 0, stream, ws);
  hipLaunchKernelGGL(vmd_final, dim3(1),  dim3(32),  0, stream, ws, out);
}